// HeatmapLoss_45260365365639
// MI455X (gfx1250) — compile-verified
//
#include <hip/hip_runtime.h>
#include <hip/hip_bf16.h>

typedef float v2f __attribute__((ext_vector_type(2)));
typedef float v8f __attribute__((ext_vector_type(8)));

#define RSQRT_2PI 0.39894228040143267794f
#define BBOX_OFFSET 64.0f
#define SIGMA 0.2f

// One block per sample. Streams the SxS cls map once (coalesced), stages it in
// LDS (pitch P=S+1, zero-padded edge row/col), computes sumC, sumC2, sum(gx^2),
// sum(gy^2) via wave reductions, then wave 0 computes the bilinear term
// t = gy^T * C * gx with V_WMMA_F32_16X16X4_F32 matvec tiles (full fp32).
template <int S>
__global__ void __launch_bounds__(256)
heatmap_level_kernel(const float* __restrict__ cls,
                     const float* __restrict__ gt,
                     float* __restrict__ partial,
                     float inv_stride, int partial_off) {
    constexpr int P  = S + 1;     // 64 or 32: multiple of 16 and 4
    constexpr int SS = S * S;
    static_assert(P % 16 == 0, "pitch must be multiple of 16");

    __shared__ float scls[P * P];
    __shared__ float sgx[P];
    __shared__ float sgy[P];
    __shared__ float red[8 * 4];

    const int tid = threadIdx.x;
    const int n   = blockIdx.x;

    // --- Gaussian parameters from gt_bbox (all threads, redundant scalar math)
    const float b0 = gt[4 * n + 0] - BBOX_OFFSET;
    const float b1 = gt[4 * n + 1] - BBOX_OFFSET;
    const float b2 = gt[4 * n + 2] - BBOX_OFFSET;
    const float b3 = gt[4 * n + 3] - BBOX_OFFSET;
    const float mu_x  = 0.5f * (b0 + b2) * inv_stride;
    const float mu_y  = 0.5f * (b1 + b3) * inv_stride;
    const float sig_x = SIGMA * (b2 - b0) * inv_stride;
    const float sig_y = SIGMA * (b3 - b1) * inv_stride;
    const float ax = RSQRT_2PI / sig_x;
    const float ay = RSQRT_2PI / sig_y;

    // --- Fill 1D gaussians (thread j handles coord j), accumulate sum of squares
    float gx2 = 0.0f, gy2 = 0.0f;
    if (tid < P) {
        float gx = 0.0f, gy = 0.0f;
        if (tid < S) {
            const float zx = (float(tid) - mu_x) / sig_x;
            const float zy = (float(tid) - mu_y) / sig_y;
            gx = ax * __expf(-0.5f * zx * zx);
            gy = ay * __expf(-0.5f * zy * zy);
        }
        sgx[tid] = gx;
        sgy[tid] = gy;
        gx2 = gx * gx;
        gy2 = gy * gy;
        // zero the padded column S (row tid) and padded row S (col tid)
        scls[tid * P + S] = 0.0f;
        if (tid < S) scls[S * P + tid] = 0.0f;
    }

    // --- Stream cls once: global (coalesced) -> LDS (pitched), reduce sums
    float sumC = 0.0f, sumC2 = 0.0f;
    const float* src = cls + (size_t)n * SS;
    for (int idx = tid; idx < SS; idx += 256) {
        const float v = src[idx];
        const int r = idx / S;           // S is a compile-time constant
        const int c = idx - r * S;
        scls[r * P + c] = v;
        sumC  += v;
        sumC2 += v * v;
    }
    __syncthreads();

    // --- Wave-level reductions (wave32), deterministic order
    #pragma unroll
    for (int o = 16; o > 0; o >>= 1) {
        sumC  += __shfl_xor(sumC,  o, 32);
        sumC2 += __shfl_xor(sumC2, o, 32);
        gx2   += __shfl_xor(gx2,   o, 32);
        gy2   += __shfl_xor(gy2,   o, 32);
    }
    const int wave = tid >> 5;
    if ((tid & 31) == 0) {
        red[wave * 4 + 0] = sumC;
        red[wave * 4 + 1] = sumC2;
        red[wave * 4 + 2] = gx2;
        red[wave * 4 + 3] = gy2;
    }
    __syncthreads();

    // --- Wave 0: t = gy^T * (C * gx) via fp32 WMMA matvec tiles
    if (tid < 32) {                      // wave-uniform branch: EXEC all-1s
        const int hi = tid >> 4;         // lane half selects K offset
        const int r  = tid & 15;         // lane-within-half = matrix row M
        float tpart = 0.0f;
        constexpr int MT = P / 16;
        constexpr int KT = P / 4;
        for (int mt = 0; mt < MT; ++mt) {
            const int m0 = mt * 16;
            v8f acc = {0.0f, 0.0f, 0.0f, 0.0f, 0.0f, 0.0f, 0.0f, 0.0f};
            #pragma unroll
            for (int kt = 0; kt < KT; ++kt) {
                const int kb = kt * 4 + 2 * hi;
                v2f a, b;
                a.x = scls[(m0 + r) * P + kb];
                a.y = scls[(m0 + r) * P + kb + 1];
                b.x = sgx[kb];            // B columns all identical = gx slice
                b.y = sgx[kb + 1];
                acc = __builtin_amdgcn_wmma_f32_16x16x4_f32(
                    false, a, false, b, (short)0, acc, false, false);
            }
            // D layout: vgpr v, lanes 0-15 -> M=m0+v, lanes 16-31 -> M=m0+8+v
            float s = 0.0f;
            #pragma unroll
            for (int v = 0; v < 8; ++v)
                s += acc[v] * sgy[m0 + v + 8 * hi];
            tpart += s;
        }
        // lanes 0..15 hold rows [m0..m0+7] part, lanes 16..31 rows [m0+8..m0+15]
        const float t = __shfl(tpart, 0, 32) + __shfl(tpart, 16, 32);

        if (tid == 0) {
            float sC = 0.0f, sC2 = 0.0f, sx2 = 0.0f, sy2 = 0.0f;
            #pragma unroll
            for (int w = 0; w < 8; ++w) {
                sC  += red[w * 4 + 0];
                sC2 += red[w * 4 + 1];
                sx2 += red[w * 4 + 2];
                sy2 += red[w * 4 + 3];
            }
            const float inv = 1.0f / sC;
            const float loss =
                (sC2 * inv * inv - 2.0f * t * inv + sx2 * sy2) / (float)SS;
            partial[partial_off + n] = loss;
        }
    }
}

// Deterministic single-block sum of the per-sample partials.
__global__ void __launch_bounds__(256)
final_reduce_kernel(const float* __restrict__ partial, float* __restrict__ out,
                    int n) {
    __shared__ float lds[256];
    const int tid = threadIdx.x;
    float s = 0.0f;
    for (int i = tid; i < n; i += 256) s += partial[i];  // fixed order per lane
    lds[tid] = s;
    __syncthreads();
    #pragma unroll
    for (int o = 128; o > 0; o >>= 1) {
        if (tid < o) lds[tid] += lds[tid + o];
        __syncthreads();
    }
    if (tid == 0) out[0] = lds[0];
}

extern "C" void kernel_launch(void* const* d_in, const int* in_sizes, int n_in,
                              void* d_out, int out_size, void* d_ws, size_t ws_size,
                              hipStream_t stream) {
    const float* cls0 = (const float*)d_in[0];
    // d_in[1] = reg0 : unused by the loss (skip 260 MB of traffic)
    const float* cls1 = (const float*)d_in[2];
    // d_in[3] = reg1 : unused
    const float* gt   = (const float*)d_in[4];
    const int N = in_sizes[4] / 4;

    float* partial = (float*)d_ws;   // 2*N floats of scratch

    heatmap_level_kernel<63><<<N, 256, 0, stream>>>(cls0, gt, partial, 0.25f, 0);
    heatmap_level_kernel<31><<<N, 256, 0, stream>>>(cls1, gt, partial, 0.125f, N);
    final_reduce_kernel<<<1, 256, 0, stream>>>(partial, (float*)d_out, 2 * N);
}